// SearchedGCNConv_14370960573129
// MI455X (gfx1250) — compile-verified
//
#include <hip/hip_runtime.h>
#include <hip/hip_bf16.h>

#define V_NODES   50000
#define E_EDGES   800000
#define E_HALF    400000
#define D         128
#define NUM_REL   200
#define BN_EPS    1e-5f
#define RT        5            // row tiles per wave (80 rows per block)
#define ROWS_BLK  (RT * 16)    // 80; 50000 / 80 = 625 exact
#define LDS_STRIDE 130         // 128 + 2 pad -> conflict-free 16-row strided reads

typedef __attribute__((ext_vector_type(2))) float v2f;
typedef __attribute__((ext_vector_type(8))) float v8f;

__device__ __forceinline__ v8f wmma4(v2f a, v2f b, v8f c) {
    return __builtin_amdgcn_wmma_f32_16x16x4_f32(
        false, a, false, b, (short)0, c, false, false);
}

// ---------------------------------------------------------------------------
// Helpers
// ---------------------------------------------------------------------------
__global__ void zero_f32(float* __restrict__ p, size_t n) {
    size_t i = (size_t)blockIdx.x * blockDim.x + threadIdx.x;
    if (i < n) p[i] = 0.0f;
}

// loop_w_scaled[k][n] = loop_rel[k] * loop_w[k][n]
__global__ void scale_loop_w(const float* __restrict__ loop_rel,
                             const float* __restrict__ loop_w,
                             float* __restrict__ outw) {
    int i = blockIdx.x * blockDim.x + threadIdx.x;
    if (i < D * D) {
        int k = i >> 7;
        outw[i] = loop_w[i] * loop_rel[k];
    }
}

// ---------------------------------------------------------------------------
// Edge scatter: acc_dir[dst] += norm * (x[src] * rel[etype])
// One wave per edge, float4 per lane (32*4 = 128 channels).
// ---------------------------------------------------------------------------
__global__ void scatter_edges(const float* __restrict__ x,
                              const float* __restrict__ rel,
                              const float* __restrict__ edge_norm,
                              const int*   __restrict__ etype,
                              const int*   __restrict__ src,
                              const int*   __restrict__ dst,
                              float* __restrict__ acc_in,
                              float* __restrict__ acc_out) {
    int e = blockIdx.x * (blockDim.x >> 5) + (threadIdx.x >> 5);
    if (e >= E_EDGES) return;
    int lane = threadIdx.x & 31;

    int   s = src[e];
    int   d = dst[e];
    int   t = etype[e];
    float nrm = edge_norm[e];
    float* acc = (e < E_HALF) ? acc_in : acc_out;   // uniform per wave

    const float4* xv4 = (const float4*)(x   + (size_t)s * D);
    const float4* rv4 = (const float4*)(rel + (size_t)t * D);
    float4 xv = xv4[lane];
    float4 rv = rv4[lane];

    float* o = acc + (size_t)d * D + lane * 4;
    atomicAdd(o + 0, xv.x * rv.x * nrm);
    atomicAdd(o + 1, xv.y * rv.y * nrm);
    atomicAdd(o + 2, xv.z * rv.z * nrm);
    atomicAdd(o + 3, xv.w * rv.w * nrm);
}

// ---------------------------------------------------------------------------
// Node GEMM: h = (acc_in@in_w + acc_out@out_w + x@loop_w')/3 + bias
// Block = 256 thr = 8 waves. Wave w owns cols [16w,16w+16), RT=5 row tiles
// (80 rows). A panel staged in LDS (one matrix at a time, 80x130 floats).
// B fragment loaded once per k-step, reused by all 5 WMMAs.
// The three (A,W) passes are source-unrolled (lambda with concrete global
// pointers) so W loads stay GLOBAL_LOAD (not FLAT) and only LDS traffic
// touches DScnt.
// Fused epilogue accumulates per-channel sum / sumsq for BatchNorm.
// Grid = 50000/80 = 625.
// ---------------------------------------------------------------------------
__global__ __launch_bounds__(256)
void gemm_nodes(const float* __restrict__ acc_in,
                const float* __restrict__ acc_out,
                const float* __restrict__ x,
                const float* __restrict__ in_w,
                const float* __restrict__ out_w,
                const float* __restrict__ loop_w_s,
                const float* __restrict__ bias,
                float* __restrict__ h_out,
                float* __restrict__ gsum,
                float* __restrict__ gsumsq) {
    __shared__ float lds[ROWS_BLK][LDS_STRIDE];   // 41,600 B

    const int lane = threadIdx.x & 31;
    const int wave = threadIdx.x >> 5;
    const int r  = lane & 15;
    const int hh = lane >> 4;
    const int row0 = blockIdx.x * ROWS_BLK;
    const int col_base = wave * 16;
    const int col = col_base + r;

    v8f c[RT] = {};

    auto panel = [&](const float* __restrict__ Am,
                     const float* __restrict__ W,
                     bool first) {
        if (!first) __syncthreads();              // WAR on LDS re-stage
        // Cooperative stage of 80x128 A panel (float2 coalesced, 20 iters/thread)
        for (int i = threadIdx.x; i < ROWS_BLK * (D / 2); i += 256) {
            int row = i >> 6;
            int c2  = i & 63;
            float2 v = ((const float2*)(Am + (size_t)(row0 + row) * D))[c2];
            *(float2*)&lds[row][c2 * 2] = v;
        }
        __syncthreads();

#pragma unroll
        for (int kk = 0; kk < 32; ++kk) {
            int k = kk * 4 + 2 * hh;              // this lane's K pair
            v2f b;
            b.x = W[(size_t)k       * D + col];
            b.y = W[(size_t)(k + 1) * D + col];
#pragma unroll
            for (int rt = 0; rt < RT; ++rt) {
                float2 av = *(const float2*)&lds[rt * 16 + r][k];  // ds_load_b64
                v2f a; a.x = av.x; a.y = av.y;
                c[rt] = wmma4(a, b, c[rt]);
            }
        }
    };

    panel(acc_in,  in_w,     true);
    panel(acc_out, out_w,    false);
    panel(x,       loop_w_s, false);

    // Epilogue: D tile layout -> VGPR j: row = base + j + 8*hh, col = col
    const float bv = bias[col];
    float lsum = 0.0f, lsq = 0.0f;
#pragma unroll
    for (int rt = 0; rt < RT; ++rt) {
#pragma unroll
        for (int j = 0; j < 8; ++j) {
            int row = row0 + rt * 16 + j + 8 * hh;
            float h = c[rt][j] * (1.0f / 3.0f) + bv;
            h_out[(size_t)row * D + col] = h;
            lsum += h;
            lsq  += h * h;
        }
    }
    atomicAdd(&gsum[col],   lsum);
    atomicAdd(&gsumsq[col], lsq);
}

// ---------------------------------------------------------------------------
// BatchNorm (training stats) + ReLU, in place over h = d_out[0 : V*D]
// ---------------------------------------------------------------------------
__global__ void bn_relu(float* __restrict__ h,
                        const float* __restrict__ gsum,
                        const float* __restrict__ gsumsq,
                        const float* __restrict__ gamma,
                        const float* __restrict__ beta,
                        int n) {
    int i = blockIdx.x * blockDim.x + threadIdx.x;
    if (i >= n) return;
    int col = i & (D - 1);
    float mean = gsum[col]   * (1.0f / (float)V_NODES);
    float var  = gsumsq[col] * (1.0f / (float)V_NODES) - mean * mean;
    float v = (h[i] - mean) * rsqrtf(var + BN_EPS) * gamma[col] + beta[col];
    h[i] = fmaxf(v, 0.0f);
}

// ---------------------------------------------------------------------------
// out_rel = rel @ w_rel   (200 x 128 @ 128 x 128), WMMA, 13 row tiles.
// Bounds handled with value selects so EXEC stays all-ones for WMMA.
// ---------------------------------------------------------------------------
__global__ void gemm_rel(const float* __restrict__ rel,
                         const float* __restrict__ w_rel,
                         float* __restrict__ out) {
    const int lane = threadIdx.x & 31;
    const int wave = threadIdx.x >> 5;
    const int r  = lane & 15;
    const int hh = lane >> 4;
    const int row_base = blockIdx.x * 16;
    const int col_base = wave * 16;

    int arow_idx = row_base + r;
    int arow_clamped = arow_idx < NUM_REL ? arow_idx : (NUM_REL - 1);
    float ascale = (arow_idx < NUM_REL) ? 1.0f : 0.0f;
    const float* arow = rel + (size_t)arow_clamped * D;

    v8f c = {};
#pragma unroll
    for (int kk = 0; kk < 32; ++kk) {
        int k = kk * 4 + 2 * hh;
        v2f a, b;
        a.x = arow[k]     * ascale;
        a.y = arow[k + 1] * ascale;
        b.x = w_rel[(size_t)k       * D + col_base + r];
        b.y = w_rel[(size_t)(k + 1) * D + col_base + r];
        c = wmma4(a, b, c);
    }

    const int col = col_base + r;
#pragma unroll
    for (int j = 0; j < 8; ++j) {
        int row = row_base + j + 8 * hh;
        if (row < NUM_REL)
            out[(size_t)row * D + col] = c[j];
    }
}

// ---------------------------------------------------------------------------
// Launch
// ---------------------------------------------------------------------------
extern "C" void kernel_launch(void* const* d_in, const int* in_sizes, int n_in,
                              void* d_out, int out_size, void* d_ws, size_t ws_size,
                              hipStream_t stream) {
    const float* x        = (const float*)d_in[0];
    const float* rel      = (const float*)d_in[1];
    const float* e_norm   = (const float*)d_in[2];
    const float* in_w     = (const float*)d_in[3];
    const float* out_w    = (const float*)d_in[4];
    const float* loop_w   = (const float*)d_in[5];
    const float* w_rel    = (const float*)d_in[6];
    const float* loop_rel = (const float*)d_in[7];
    const float* bias     = (const float*)d_in[8];
    const float* bn_gamma = (const float*)d_in[9];
    const float* bn_beta  = (const float*)d_in[10];
    const int*   etype    = (const int*)d_in[11];
    const int*   src      = (const int*)d_in[12];
    const int*   dst      = (const int*)d_in[13];

    float* out_x   = (float*)d_out;                       // [V, 128]
    float* out_rel = out_x + (size_t)V_NODES * D;         // [200, 128]

    // Workspace layout (floats)
    float* ws      = (float*)d_ws;
    float* acc_in  = ws;                                  // V*128
    float* acc_out = acc_in  + (size_t)V_NODES * D;       // V*128
    float* lw_s    = acc_out + (size_t)V_NODES * D;       // 128*128
    float* gsum    = lw_s + (size_t)D * D;                // 128
    float* gsumsq  = gsum + D;                            // 128

    // 1) zero accumulators + stats
    {
        size_t nz = (size_t)2 * V_NODES * D;
        zero_f32<<<(unsigned)((nz + 255) / 256), 256, 0, stream>>>(acc_in, nz);
        zero_f32<<<1, 256, 0, stream>>>(gsum, (size_t)2 * D);
    }

    // 2) fold loop_rel into loop_w
    scale_loop_w<<<(D * D + 255) / 256, 256, 0, stream>>>(loop_rel, loop_w, lw_s);

    // 3) edge scatter (one wave per edge)
    scatter_edges<<<(E_EDGES + 7) / 8, 256, 0, stream>>>(
        x, rel, e_norm, etype, src, dst, acc_in, acc_out);

    // 4) dense node GEMM (WMMA, LDS-staged A, B reuse) + fused BN stats
    gemm_nodes<<<V_NODES / ROWS_BLK, 256, 0, stream>>>(
        acc_in, acc_out, x, in_w, out_w, lw_s, bias, out_x, gsum, gsumsq);

    // 5) BatchNorm + ReLU in place
    bn_relu<<<(V_NODES * D + 255) / 256, 256, 0, stream>>>(
        out_x, gsum, gsumsq, bn_gamma, bn_beta, V_NODES * D);

    // 6) out_rel = rel @ w_rel (WMMA)
    gemm_rel<<<(NUM_REL + 15) / 16, 256, 0, stream>>>(rel, w_rel, out_rel);
}